// GCN_34102040330816
// MI455X (gfx1250) — compile-verified
//
#include <hip/hip_runtime.h>
#include <hip/hip_bf16.h>

typedef __attribute__((ext_vector_type(2))) float v2f;
typedef __attribute__((ext_vector_type(8))) float v8f;

#define PADF 48  // padded feature stride for all intermediate [N, F] buffers

__device__ __forceinline__ float selu_f(float x) {
  const float alpha = 1.6732632423543772848f;
  const float scale = 1.0507009873554804934f;
  return scale * (x > 0.0f ? x : alpha * (expf(x) - 1.0f));
}

__global__ void k_fill(float* __restrict__ p, float v, int n) {
  int i = blockIdx.x * blockDim.x + threadIdx.x;
  if (i < n) p[i] = v;
}

// deg[dst] += 1 over all edges (deg pre-filled with 1.0 for the self loop)
__global__ void k_degree(const int* __restrict__ dst, float* __restrict__ deg, int E) {
  int e = blockIdx.x * blockDim.x + threadIdx.x;
  if (e < E)
    __hip_atomic_fetch_add(&deg[dst[e]], 1.0f, __ATOMIC_RELAXED, __HIP_MEMORY_SCOPE_AGENT);
}

__global__ void k_rsqrt(const float* __restrict__ deg, float* __restrict__ dinv, int n) {
  int i = blockIdx.x * blockDim.x + threadIdx.x;
  if (i < n) {
    float d = deg[i];
    dinv[i] = d > 0.0f ? rsqrtf(d) : 0.0f;
  }
}

// per-edge normalization coefficient, reused by all 4 layers
__global__ void k_norm(const int* __restrict__ src, const int* __restrict__ dst,
                       const float* __restrict__ dinv, float* __restrict__ norm, int E) {
  int e = blockIdx.x * blockDim.x + threadIdx.x;
  if (e < E) norm[e] = dinv[src[e]] * dinv[dst[e]];
}

// H[N, PADF] = A[N, lda(:K)] @ W[K, FOUT]; fp32 WMMA 16x16x4, one wave per 16x16 tile.
// Requires N % 16 == 0 (grid.x = N/16). All loads are clamped in-bounds + selected,
// so codegen is branch-free (no exec-mask churn around the wmma chain).
template <int K, int FOUT>
__global__ void k_gemm_wmma(const float* __restrict__ A, int lda,
                            const float* __restrict__ W,
                            float* __restrict__ H) {
  const int lane = threadIdx.x & 31;
  const int half = lane >> 4;    // 0: K pair {+0,+1}; 1: K pair {+2,+3}
  const int lid  = lane & 15;
  const int row  = blockIdx.x * 16 + lid;
  const int col  = blockIdx.y * 16 + lid;
  const bool cv  = (FOUT % 16 == 0) || (col < FOUT);
  const int  cc  = cv ? col : (FOUT - 1);  // clamped column (loads always in-bounds)

  const float* Ap = A + row * lda + 2 * half;
  v8f acc = {};
  constexpr int K4 = K & ~3;
#pragma unroll 4
  for (int k0 = 0; k0 < K4; k0 += 4) {
    const int kb = k0 + 2 * half;
    v2f a, b;
    a.x = Ap[k0];          // pair load -> global_load_b64
    a.y = Ap[k0 + 1];
    float b0 = W[kb * FOUT + cc];
    float b1 = W[(kb + 1) * FOUT + cc];
    b.x = cv ? b0 : 0.0f;
    b.y = cv ? b1 : 0.0f;
    acc = __builtin_amdgcn_wmma_f32_16x16x4_f32(false, a, false, b, (short)0, acc,
                                                false, false);
  }
  if constexpr (K % 4 != 0) {  // compile-time K tail: clamp + select, still branch-free
    const int kb  = K4 + 2 * half;
    const int k0c = (kb     < K) ? kb     : (K - 1);
    const int k1c = (kb + 1 < K) ? kb + 1 : (K - 1);
    v2f a, b;
    float a0 = A[row * lda + k0c];
    float a1 = A[row * lda + k1c];
    a.x = (kb     < K) ? a0 : 0.0f;
    a.y = (kb + 1 < K) ? a1 : 0.0f;
    float b0 = W[k0c * FOUT + cc];
    float b1 = W[k1c * FOUT + cc];
    b.x = (cv && kb     < K) ? b0 : 0.0f;
    b.y = (cv && kb + 1 < K) ? b1 : 0.0f;
    acc = __builtin_amdgcn_wmma_f32_16x16x4_f32(false, a, false, b, (short)0, acc,
                                                false, false);
  }
  // C/D layout: VGPR r -> row (r + 8*half), lane -> column
  const int hbase = (blockIdx.x * 16 + 8 * half) * PADF + blockIdx.y * 16 + lid;
#pragma unroll
  for (int r = 0; r < 8; ++r)
    H[hbase + r * PADF] = acc[r];
}

// agg[dst, f] += norm[e] * h[src, f]; one thread per (edge, feature).
// F is compile-time -> t/F becomes a magic multiply (hot kernel, keep VALU lean).
template <int F>
__global__ void k_scatter(const float* __restrict__ h, const int* __restrict__ src,
                          const int* __restrict__ dst, const float* __restrict__ norm,
                          float* __restrict__ agg, int E) {
  int t = blockIdx.x * blockDim.x + threadIdx.x;
  if (t >= E * F) return;
  int e = t / F;
  int f = t - e * F;
  float v = norm[e] * h[src[e] * PADF + f];
  __hip_atomic_fetch_add(&agg[dst[e] * PADF + f], v,
                         __ATOMIC_RELAXED, __HIP_MEMORY_SCOPE_AGENT);
}

// out = selu(agg + dinv^2 * h + bias)
template <int F>
__global__ void k_finalize(const float* __restrict__ h, const float* __restrict__ agg,
                           const float* __restrict__ dinv, const float* __restrict__ bias,
                           float* __restrict__ out, int ldo, int N) {
  int t = blockIdx.x * blockDim.x + threadIdx.x;
  if (t >= N * F) return;
  int n = t / F;
  int f = t - n * F;
  float di = dinv[n];
  float v = agg[n * PADF + f] + di * di * h[n * PADF + f] + bias[f];
  out[n * ldo + f] = selu_f(v);
}

extern "C" void kernel_launch(void* const* d_in, const int* in_sizes, int n_in,
                              void* d_out, int out_size, void* d_ws, size_t ws_size,
                              hipStream_t stream) {
  const float* x  = (const float*)d_in[0];
  const int*   ei = (const int*)d_in[1];
  const float* W1 = (const float*)d_in[2];
  const float* b1 = (const float*)d_in[3];
  const float* W2 = (const float*)d_in[4];
  const float* b2 = (const float*)d_in[5];
  const float* W3 = (const float*)d_in[6];
  const float* b3 = (const float*)d_in[7];
  const float* W4 = (const float*)d_in[8];
  const float* b4 = (const float*)d_in[9];

  const int E = in_sizes[1] / 2;
  const int N = in_sizes[0] / 128;  // 100000, multiple of 16
  const int* src = ei;
  const int* dst = ei + E;

  float* ws   = (float*)d_ws;
  float* deg  = ws;                    // N
  float* dinv = deg  + N;              // N
  float* norm = dinv + N;              // E
  float* hG   = norm + E;              // N*PADF : GEMM output of current layer
  float* agg  = hG   + (size_t)N * PADF;
  float* hIn  = agg  + (size_t)N * PADF;

  auto cdiv = [](long long a, long long b) { return (int)((a + b - 1) / b); };
  const int NT = N / 16;

  // Degree / normalization pipeline (cheap, recomputed every call)
  k_fill  <<<cdiv(N, 256), 256, 0, stream>>>(deg, 1.0f, N);
  k_degree<<<cdiv(E, 256), 256, 0, stream>>>(dst, deg, E);
  k_rsqrt <<<cdiv(N, 256), 256, 0, stream>>>(deg, dinv, N);
  k_norm  <<<cdiv(E, 256), 256, 0, stream>>>(src, dst, dinv, norm, E);

  // ---- Layer 1: 128 -> 15 ----
  k_gemm_wmma<128, 15><<<dim3(NT, 1), 32, 0, stream>>>(x, 128, W1, hG);
  k_fill<<<cdiv(N * PADF, 256), 256, 0, stream>>>(agg, 0.0f, N * PADF);
  k_scatter<15><<<cdiv((long long)E * 15, 256), 256, 0, stream>>>(hG, src, dst, norm, agg, E);
  k_finalize<15><<<cdiv((long long)N * 15, 256), 256, 0, stream>>>(hG, agg, dinv, b1, hIn, PADF, N);

  // ---- Layer 2: 15 -> 20 ----
  k_gemm_wmma<15, 20><<<dim3(NT, 2), 32, 0, stream>>>(hIn, PADF, W2, hG);
  k_fill<<<cdiv(N * PADF, 256), 256, 0, stream>>>(agg, 0.0f, N * PADF);
  k_scatter<20><<<cdiv((long long)E * 20, 256), 256, 0, stream>>>(hG, src, dst, norm, agg, E);
  k_finalize<20><<<cdiv((long long)N * 20, 256), 256, 0, stream>>>(hG, agg, dinv, b2, hIn, PADF, N);

  // ---- Layer 3: 20 -> 27 ----
  k_gemm_wmma<20, 27><<<dim3(NT, 2), 32, 0, stream>>>(hIn, PADF, W3, hG);
  k_fill<<<cdiv(N * PADF, 256), 256, 0, stream>>>(agg, 0.0f, N * PADF);
  k_scatter<27><<<cdiv((long long)E * 27, 256), 256, 0, stream>>>(hG, src, dst, norm, agg, E);
  k_finalize<27><<<cdiv((long long)N * 27, 256), 256, 0, stream>>>(hG, agg, dinv, b3, hIn, PADF, N);

  // ---- Layer 4: 27 -> 36 (output, dense stride 36) ----
  k_gemm_wmma<27, 36><<<dim3(NT, 3), 32, 0, stream>>>(hIn, PADF, W4, hG);
  k_fill<<<cdiv(N * PADF, 256), 256, 0, stream>>>(agg, 0.0f, N * PADF);
  k_scatter<36><<<cdiv((long long)E * 36, 256), 256, 0, stream>>>(hG, src, dst, norm, agg, E);
  k_finalize<36><<<cdiv((long long)N * 36, 256), 256, 0, stream>>>(hG, agg, dinv, b4,
                                                                   (float*)d_out, 36, N);
}